// ECNetWrapperGNN_12601434046587
// MI455X (gfx1250) — compile-verified
//
#include <hip/hip_runtime.h>

#define N_NODES  100000
#define N_EDGES  1600000
#define N_GRAPHS 250
#define POOL_CHUNK 16

typedef float v2f __attribute__((ext_vector_type(2)));
typedef float v8f __attribute__((ext_vector_type(8)));

// ---------- small prep kernels ----------

__global__ void pad_w0(const float* __restrict__ Wl, const float* __restrict__ Wr,
                       float* __restrict__ Wlp, float* __restrict__ Wrp) {
    int idx = blockIdx.x * blockDim.x + threadIdx.x;   // 64 x 16
    if (idx >= 64 * 16) return;
    int o = idx >> 4, k = idx & 15;
    Wlp[idx] = (k < 13) ? Wl[o * 13 + k] : 0.0f;
    Wrp[idx] = (k < 13) ? Wr[o * 13 + k] : 0.0f;
}

__global__ void build_x0(const float* __restrict__ node_h, const float* __restrict__ glob,
                         const int* __restrict__ batch, float* __restrict__ x0) {
    int idx = blockIdx.x * blockDim.x + threadIdx.x;   // N * 16
    if (idx >= N_NODES * 16) return;
    int n = idx >> 4, c = idx & 15;
    float v = 0.0f;
    if (c < 9)       v = node_h[n * 9 + c];
    else if (c < 13) v = glob[batch[n] * 4 + (c - 9)];
    x0[idx] = v;
}

__global__ void edge_degree(const int* __restrict__ dst, float* __restrict__ deg) {
    int e = blockIdx.x * blockDim.x + threadIdx.x;
    if (e >= N_EDGES) return;
    atomicAdd(deg + dst[e], 1.0f);
}

// ---------- edge scatter (mean-agg numerator): L2-resident f32 atomics ----------
// LOG_TPE: threads per edge = 2^LOG_TPE, each thread moves a float4. F = 4<<LOG_TPE.
template<int LOG_TPE>
__global__ void edge_scatter(const int* __restrict__ src, const int* __restrict__ dst,
                             const float* __restrict__ x, float* __restrict__ agg) {
    const int F = 4 << LOG_TPE;
    int idx = blockIdx.x * blockDim.x + threadIdx.x;
    int e = idx >> LOG_TPE;
    if (e >= N_EDGES) return;
    int f = (idx & ((1 << LOG_TPE) - 1)) << 2;
    int s = src[e], d = dst[e];
    float4 v = *(const float4*)(x + (size_t)s * F + f);
    float* o = agg + (size_t)d * F + f;
    atomicAdd(o + 0, v.x);
    atomicAdd(o + 1, v.y);
    atomicAdd(o + 2, v.z);
    atomicAdd(o + 3, v.w);
}

// ---------- fused SAGE linear: out = act( (agg/deg) @ Wl^T + x @ Wr^T + b ) ----------
// Block = 256 threads = 8 waves. Stage both 32xK A tiles (mean-agg, x) in LDS
// once (coalesced float4), then each wave computes a 16x16 tile via
// V_WMMA_F32_16X16X4_F32 fed from ds_load. LDS row stride padded to K+2 so the
// 16 lanes reading one column land on distinct banks.
template<int K, int RELU>
__global__ __launch_bounds__(256) void sage_gemm(
    const float* __restrict__ x, const float* __restrict__ agg,
    const float* __restrict__ deg,
    const float* __restrict__ Wl, const float* __restrict__ Wr,
    const float* __restrict__ bias, float* __restrict__ out) {
    constexpr int LDK = K + 2;
    __shared__ float smean[32 * LDK];
    __shared__ float sx[32 * LDK];
    __shared__ float sinv[32];

    const int tid = threadIdx.x;
    const int rowBlock = blockIdx.x * 32;

    if (tid < 32) sinv[tid] = 1.0f / fmaxf(deg[rowBlock + tid], 1.0f);
    __syncthreads();

    // stage 32 rows x K cols of both tensors; 8*K float4s total
    for (int i = tid; i < 8 * K; i += 256) {
        int r  = i / (K / 4);
        int c4 = (i % (K / 4)) * 4;
        const float4 va = *(const float4*)(agg + (size_t)(rowBlock + r) * K + c4);
        const float4 vx = *(const float4*)(x   + (size_t)(rowBlock + r) * K + c4);
        const float inv = sinv[r];
        float* pm = smean + r * LDK + c4;
        pm[0] = va.x * inv; pm[1] = va.y * inv; pm[2] = va.z * inv; pm[3] = va.w * inv;
        float* px = sx + r * LDK + c4;
        px[0] = vx.x; px[1] = vx.y; px[2] = vx.z; px[3] = vx.w;
    }
    __syncthreads();

    const int lane = tid & 31;
    const int wave = tid >> 5;
    const int rtile   = wave >> 2;            // 0/1: row sub-tile
    const int colBase = (wave & 3) * 16;      // 4 col tiles cover 64 outputs
    const int half = lane >> 4;               // K-half select (ISA 16x4 A layout)
    const int l15  = lane & 15;
    const int lrow = rtile * 16 + l15;        // local A row (lane -> M)
    const int bcol = colBase + l15;           // B/C/D lane -> N

    const float* mrow  = smean + lrow * LDK;
    const float* xrow  = sx    + lrow * LDK;
    const float* WlRow = Wl + (size_t)bcol * K;   // B[k][n] = W[n][k]
    const float* WrRow = Wr + (size_t)bcol * K;

    v8f c = {};
    #pragma unroll
    for (int k = 0; k < K; k += 4) {
        int kk = k + half * 2;
        v2f a, b;
        a.x = mrow[kk];  a.y = mrow[kk + 1];
        b.x = WlRow[kk]; b.y = WlRow[kk + 1];
        c = __builtin_amdgcn_wmma_f32_16x16x4_f32(false, a, false, b, (short)0, c, false, false);
        a.x = xrow[kk];  a.y = xrow[kk + 1];
        b.x = WrRow[kk]; b.y = WrRow[kk + 1];
        c = __builtin_amdgcn_wmma_f32_16x16x4_f32(false, a, false, b, (short)0, c, false, false);
    }
    const float bb = bias[bcol];
    const int rowBase = rowBlock + rtile * 16;
    #pragma unroll
    for (int j = 0; j < 8; ++j) {   // VGPR j -> row M = j (+8 for upper lanes)
        float val = c[j] + bb;
        if (RELU) val = fmaxf(val, 0.0f);
        out[(size_t)(rowBase + j + half * 8) * 64 + bcol] = val;
    }
}

// ---------- graph mean pooling (batch_idx is sorted -> segmented accumulate) ----------

__global__ void pool_cnt(const int* __restrict__ batch, float* __restrict__ cnt) {
    int chunk = blockIdx.x * blockDim.x + threadIdx.x;
    if (chunk >= N_NODES / POOL_CHUNK) return;
    int n0 = chunk * POOL_CHUNK;
    float acc = 0.0f;
    int cur = batch[n0];
    for (int i = 0; i < POOL_CHUNK; ++i) {
        int b = batch[n0 + i];
        if (b != cur) { atomicAdd(cnt + cur, acc); acc = 0.0f; cur = b; }
        acc += 1.0f;
    }
    atomicAdd(cnt + cur, acc);
}

__global__ void pool_sum(const int* __restrict__ batch, const float* __restrict__ x,
                         float* __restrict__ pooled) {
    int idx = blockIdx.x * blockDim.x + threadIdx.x;   // (N/16) * 64
    if (idx >= (N_NODES / POOL_CHUNK) * 64) return;
    int f = idx & 63;
    int chunk = idx >> 6;
    int n0 = chunk * POOL_CHUNK;
    float acc = 0.0f;
    int cur = batch[n0];
    for (int i = 0; i < POOL_CHUNK; ++i) {
        int n = n0 + i;
        int b = batch[n];
        if (b != cur) { atomicAdd(pooled + cur * 64 + f, acc); acc = 0.0f; cur = b; }
        acc += x[(size_t)n * 64 + f];
    }
    atomicAdd(pooled + cur * 64 + f, acc);
}

// ---------- tiny MLP head (250x64 — negligible cost) ----------

__global__ void mlp_layer(const float* __restrict__ in, const float* __restrict__ cnt,
                          const float* __restrict__ W, const float* __restrict__ b,
                          float* __restrict__ out, int relu) {
    int g = blockIdx.x;          // graph
    int t = threadIdx.x;         // output feature (64)
    float inv = 1.0f;
    if (cnt) inv = 1.0f / fmaxf(cnt[g], 1.0f);
    const float* row = in + g * 64;
    const float* w   = W + t * 64;
    float acc = b[t];
    #pragma unroll 8
    for (int k = 0; k < 64; ++k) acc += (row[k] * inv) * w[k];
    if (relu) acc = fmaxf(acc, 0.0f);
    out[g * 64 + t] = acc;
}

__global__ void mlp_final(const float* __restrict__ in, const float* __restrict__ W,
                          const float* __restrict__ b, float* __restrict__ out) {
    int g = blockIdx.x * blockDim.x + threadIdx.x;
    if (g >= N_GRAPHS) return;
    const float* row = in + g * 64;
    float acc = b[0];
    #pragma unroll 8
    for (int k = 0; k < 64; ++k) acc += row[k] * W[k];
    out[g] = acc;
}

// ---------- launch ----------

extern "C" void kernel_launch(void* const* d_in, const int* in_sizes, int n_in,
                              void* d_out, int out_size, void* d_ws, size_t ws_size,
                              hipStream_t stream) {
    const float* node_h = (const float*)d_in[0];
    const float* glob   = (const float*)d_in[1];
    const int*   ei     = (const int*)d_in[2];
    const int*   batch  = (const int*)d_in[3];
    const float* W0l = (const float*)d_in[4];
    const float* b0  = (const float*)d_in[5];
    const float* W0r = (const float*)d_in[6];
    const float* W1l = (const float*)d_in[7];
    const float* b1  = (const float*)d_in[8];
    const float* W1r = (const float*)d_in[9];
    const float* W2l = (const float*)d_in[10];
    const float* b2  = (const float*)d_in[11];
    const float* W2r = (const float*)d_in[12];
    const float* mW0 = (const float*)d_in[13];
    const float* mb0 = (const float*)d_in[14];
    const float* mW1 = (const float*)d_in[15];
    const float* mb1 = (const float*)d_in[16];
    const float* mW2 = (const float*)d_in[17];
    const float* mb2 = (const float*)d_in[18];
    const float* mW3 = (const float*)d_in[19];
    const float* mb3 = (const float*)d_in[20];

    const int* src  = ei;
    const int* dstp = ei + N_EDGES;

    float* ws = (float*)d_ws;
    float* x0     = ws; ws += (size_t)N_NODES * 16;
    float* agg    = ws; ws += (size_t)N_NODES * 64;
    float* h1     = ws; ws += (size_t)N_NODES * 64;
    float* h2     = ws; ws += (size_t)N_NODES * 64;
    float* deg    = ws; ws += N_NODES;
    float* pooled = ws; ws += N_GRAPHS * 64;
    float* cnt    = ws; ws += N_GRAPHS;
    float* g0     = ws; ws += N_GRAPHS * 64;
    float* g1     = ws; ws += N_GRAPHS * 64;
    float* g2     = ws; ws += N_GRAPHS * 64;
    float* W0lp   = ws; ws += 64 * 16;
    float* W0rp   = ws; ws += 64 * 16;

    const int B = 256;
    hipMemsetAsync(deg,    0, (size_t)N_NODES * sizeof(float), stream);
    hipMemsetAsync(pooled, 0, (size_t)N_GRAPHS * 64 * sizeof(float), stream);
    hipMemsetAsync(cnt,    0, (size_t)N_GRAPHS * sizeof(float), stream);

    pad_w0  <<<(64 * 16 + B - 1) / B, B, 0, stream>>>(W0l, W0r, W0lp, W0rp);
    build_x0<<<(N_NODES * 16 + B - 1) / B, B, 0, stream>>>(node_h, glob, batch, x0);
    edge_degree<<<(N_EDGES + B - 1) / B, B, 0, stream>>>(dstp, deg);

    // layer 0 (K = 16, zero-padded from 13)
    hipMemsetAsync(agg, 0, (size_t)N_NODES * 16 * sizeof(float), stream);
    edge_scatter<2><<<(N_EDGES * 4 + B - 1) / B, B, 0, stream>>>(src, dstp, x0, agg);
    sage_gemm<16, 1><<<N_NODES / 32, B, 0, stream>>>(x0, agg, deg, W0lp, W0rp, b0, h1);

    // layer 1 (K = 64)
    hipMemsetAsync(agg, 0, (size_t)N_NODES * 64 * sizeof(float), stream);
    edge_scatter<4><<<(N_EDGES * 16 + B - 1) / B, B, 0, stream>>>(src, dstp, h1, agg);
    sage_gemm<64, 1><<<N_NODES / 32, B, 0, stream>>>(h1, agg, deg, W1l, W1r, b1, h2);

    // layer 2 (K = 64, no relu)
    hipMemsetAsync(agg, 0, (size_t)N_NODES * 64 * sizeof(float), stream);
    edge_scatter<4><<<(N_EDGES * 16 + B - 1) / B, B, 0, stream>>>(src, dstp, h2, agg);
    sage_gemm<64, 0><<<N_NODES / 32, B, 0, stream>>>(h2, agg, deg, W2l, W2r, b2, h1);

    // mean pool over graphs (segmented; batch_idx sorted)
    pool_cnt<<<(N_NODES / POOL_CHUNK + B - 1) / B, B, 0, stream>>>(batch, cnt);
    pool_sum<<<((N_NODES / POOL_CHUNK) * 64 + B - 1) / B, B, 0, stream>>>(batch, h1, pooled);

    // MLP head
    mlp_layer<<<N_GRAPHS, 64, 0, stream>>>(pooled, cnt, mW0, mb0, g0, 1);
    mlp_layer<<<N_GRAPHS, 64, 0, stream>>>(g0, nullptr, mW1, mb1, g1, 1);
    mlp_layer<<<N_GRAPHS, 64, 0, stream>>>(g1, nullptr, mW2, mb2, g2, 1);
    mlp_final<<<(N_GRAPHS + 63) / 64, 64, 0, stream>>>(g2, mW3, mb3, (float*)d_out);
}